// MultiBoxLoss_40467181863560
// MI455X (gfx1250) — compile-verified
//
#include <hip/hip_runtime.h>
#include <math.h>

#define THRESHOLD_F   0.35f
#define SMOOTH_POINT_F 0.2f
#define VAR0_F 0.1f
#define VAR1_F 0.2f
#define MAX_T 64

typedef __attribute__((ext_vector_type(2))) float v2f;
typedef __attribute__((ext_vector_type(8))) float v8f;

#if defined(__has_builtin)
#if __has_builtin(__builtin_amdgcn_wmma_f32_16x16x4_f32)
#define HAVE_WMMA_F32_16X16X4 1
#endif
#endif

// Full 32-lane sum, result in every lane.
// WMMA trick: A(16x4) holds v in K=0 (lanes 0-15) / K=2 (lanes 16-31),
// B = ones => D[m,n] = v_m + v_{m+16}. Then 8 VALU adds give per-halfwave
// column sums; one xor-16 shuffle completes the reduction.
__device__ __forceinline__ float wave_sum32(float v) {
#if defined(HAVE_WMMA_F32_16X16X4)
  v2f a; a[0] = v;    a[1] = 0.0f;
  v2f b; b[0] = 1.0f; b[1] = 1.0f;
  v8f c = {0.f, 0.f, 0.f, 0.f, 0.f, 0.f, 0.f, 0.f};
  c = __builtin_amdgcn_wmma_f32_16x16x4_f32(false, a, false, b, (short)0, c,
                                            false, false);
  float t = c[0] + c[1] + c[2] + c[3] + c[4] + c[5] + c[6] + c[7];
  t += __shfl_xor(t, 16, 32);
  return t;
#else
  for (int m = 1; m < 32; m <<= 1) v += __shfl_xor(v, m, 32);
  return v;
#endif
}

__device__ __forceinline__ float smooth_l1f(float x) {
  float a = fabsf(x);
  return (a < 1.0f) ? 0.5f * a * a : a - 0.5f;
}

// ---------------- init ----------------
__global__ void k_init(float* accum, int* npos, int B) {
  int i = blockIdx.x * blockDim.x + threadIdx.x;
  if (i < 8) accum[i] = 0.0f;
  if (i < B) npos[i] = 0;
}

// ---------------- per-prior best truth (max/argmax over T) ----------------
__global__ void k_best_truth(const float* __restrict__ priors,
                             const float* __restrict__ targets,
                             float* __restrict__ bto, int* __restrict__ bti,
                             int P, int T) {
  int b = blockIdx.y;
  int p = blockIdx.x * blockDim.x + threadIdx.x;
  __shared__ float tb[MAX_T * 4];
  for (int i = threadIdx.x; i < T * 4; i += blockDim.x) {
    int t = i >> 2, cidx = i & 3;
    tb[i] = targets[((size_t)b * T + t) * 15 + cidx];
  }
  __syncthreads();
  if (p >= P) return;
  float cx = priors[p * 4 + 0], cy = priors[p * 4 + 1];
  float pw = priors[p * 4 + 2], ph = priors[p * 4 + 3];
  float px1 = cx - 0.5f * pw, py1 = cy - 0.5f * ph;
  float px2 = cx + 0.5f * pw, py2 = cy + 0.5f * ph;
  float parea = pw * ph;
  float best = -1.0f;
  int bidx = 0;
  for (int t = 0; t < T; ++t) {
    float tx1 = tb[t * 4 + 0], ty1 = tb[t * 4 + 1];
    float tx2 = tb[t * 4 + 2], ty2 = tb[t * 4 + 3];
    float lx = fmaxf(px1, tx1), ly = fmaxf(py1, ty1);
    float rx = fminf(px2, tx2), ry = fminf(py2, ty2);
    float w = fmaxf(rx - lx, 0.0f), h = fmaxf(ry - ly, 0.0f);
    float inter = w * h;
    float ta = (tx2 - tx1) * (ty2 - ty1);
    float ov = inter / (ta + parea - inter);
    if (ov > best) { best = ov; bidx = t; }   // strict > : first-max tie-break
  }
  bto[(size_t)b * P + p] = best;
  bti[(size_t)b * P + p] = bidx;
}

// ---------------- per-truth best prior (argmax over P) ----------------
__global__ void k_best_prior(const float* __restrict__ priors,
                             const float* __restrict__ targets,
                             int* __restrict__ bp_idx, int P, int T) {
  int b = blockIdx.y, t = blockIdx.x;
  float tx1 = targets[((size_t)b * T + t) * 15 + 0];
  float ty1 = targets[((size_t)b * T + t) * 15 + 1];
  float tx2 = targets[((size_t)b * T + t) * 15 + 2];
  float ty2 = targets[((size_t)b * T + t) * 15 + 3];
  float ta = (tx2 - tx1) * (ty2 - ty1);
  unsigned long long bestk = 0ull;
  for (int p = threadIdx.x; p < P; p += blockDim.x) {
    float cx = priors[p * 4 + 0], cy = priors[p * 4 + 1];
    float pw = priors[p * 4 + 2], ph = priors[p * 4 + 3];
    float px1 = cx - 0.5f * pw, py1 = cy - 0.5f * ph;
    float px2 = cx + 0.5f * pw, py2 = cy + 0.5f * ph;
    float lx = fmaxf(px1, tx1), ly = fmaxf(py1, ty1);
    float rx = fminf(px2, tx2), ry = fminf(py2, ty2);
    float w = fmaxf(rx - lx, 0.0f), h = fmaxf(ry - ly, 0.0f);
    float inter = w * h;
    float ov = inter / (ta + pw * ph - inter);     // >= 0
    unsigned long long key =
        ((unsigned long long)__float_as_uint(ov) << 32) |
        (unsigned long long)(0xFFFFFFFFu - (unsigned)p);  // min-p tie-break
    if (key > bestk) bestk = key;
  }
  __shared__ unsigned long long sm[256];
  sm[threadIdx.x] = bestk;
  __syncthreads();
  for (int s = blockDim.x >> 1; s > 0; s >>= 1) {
    if (threadIdx.x < (unsigned)s)
      if (sm[threadIdx.x + s] > sm[threadIdx.x]) sm[threadIdx.x] = sm[threadIdx.x + s];
    __syncthreads();
  }
  if (threadIdx.x == 0)
    bp_idx[b * T + t] = (int)(0xFFFFFFFFu - (unsigned)(sm[0] & 0xFFFFFFFFull));
}

// ---------------- forced-match scatter (sequential t: last write wins) -----
__global__ void k_scatter(const int* __restrict__ bp_idx, float* bto, int* bti,
                          int P, int T, int B) {
  int b = blockIdx.x * blockDim.x + threadIdx.x;
  if (b >= B) return;
  for (int t = 0; t < T; ++t) {
    int p = bp_idx[b * T + t];
    bto[(size_t)b * P + p] = 2.0f;
    bti[(size_t)b * P + p] = t;
  }
}

// ---------------- single streaming pass over loc/conf/iou -----------------
__global__ void k_loss(const float* __restrict__ loc,
                       const float* __restrict__ conf,
                       const float* __restrict__ iou,
                       const float* __restrict__ priors,
                       const float* __restrict__ targets,
                       const float* __restrict__ bto,
                       const int* __restrict__ bti,
                       float* __restrict__ loss_c,
                       float* accum, int* npos, int P, int T) {
  int b = blockIdx.y;
  int p = blockIdx.x * blockDim.x + threadIdx.x;
  __shared__ float sh[MAX_T * 15];
  for (int i = threadIdx.x; i < T * 15; i += blockDim.x)
    sh[i] = targets[(size_t)b * T * 15 + i];
  __syncthreads();

  bool valid = (p < P);
  int pi = valid ? p : 0;
  size_t bp = (size_t)b * P + pi;

  float ov = bto[bp];
  int ti = bti[bp];
  float label = sh[ti * 15 + 14];
  int conf_t = (ov < THRESHOLD_F) ? 0 : (int)label;
  conf_t = conf_t < 0 ? 0 : (conf_t > 1 ? 1 : conf_t);
  bool pos = valid && (conf_t > 0);

  float c0 = conf[bp * 2 + 0], c1 = conf[bp * 2 + 1];
  float mx = fmaxf(c0, c1);
  float lse = mx + logf(expf(c0 - mx) + expf(c1 - mx));
  float ce = lse - (conf_t ? c1 : c0);
  if (valid) loss_c[bp] = pos ? 0.0f : ce;

  float l_eiou = 0.f, l_lm = 0.f, l_iou = 0.f, l_posce = 0.f, cnt = 0.f;
  if (pos) {
    const float* tr = &sh[ti * 15];
    float cx = priors[pi * 4 + 0], cy = priors[pi * 4 + 1];
    float pw = priors[pi * 4 + 2], ph = priors[pi * 4 + 3];
    float sx = 1.0f / (VAR0_F * pw), sy = 1.0f / (VAR0_F * ph);
    // encode box
    float gx = ((tr[0] + tr[2]) * 0.5f - cx) * sx;
    float gy = ((tr[1] + tr[3]) * 0.5f - cy) * sy;
    float gw = logf((tr[2] - tr[0]) / pw) * (1.0f / VAR1_F);
    float gh = logf((tr[3] - tr[1]) / ph) * (1.0f / VAR1_F);
    const float* ld = &loc[bp * 14];
    // eiou on encoded coords (mirrors reference exactly)
    {
      float px1 = ld[0], py1 = ld[1], px2 = ld[2], py2 = ld[3];
      float ex1 = fminf(px1, gx), ey1 = fminf(py1, gy);
      float ix1 = fmaxf(px1, gx), iy1 = fmaxf(py1, gy);
      float ix2 = fminf(px2, gw), iy2 = fminf(py2, gh);
      float xmin = fminf(ix1, ix2), ymin = fminf(iy1, iy2);
      float xmax = fmaxf(ix1, ix2), ymax = fmaxf(iy1, iy2);
      float inter = (ix2 - ex1) * (iy2 - ey1) + (xmin - ex1) * (ymin - ey1)
                  - (ix1 - ex1) * (ymax - ey1) - (xmax - ex1) * (iy1 - ey1);
      float uni = (px2 - px1) * (py2 - py1) + (gw - gx) * (gh - gy)
                - inter + 1e-7f;
      float ious = 1.0f - inter / uni;
      float sgn = (ious < SMOOTH_POINT_F) ? 1.0f : 0.0f;
      l_eiou = 0.5f * sgn * ious * ious * (1.0f / SMOOTH_POINT_F)
             + (1.0f - sgn) * (ious - 0.5f * SMOOTH_POINT_F);
    }
    // landmarks
    for (int i = 0; i < 5; ++i) {
      float glx = (tr[4 + 2 * i] - cx) * sx;
      float gly = (tr[5 + 2 * i] - cy) * sy;
      l_lm += smooth_l1f(ld[4 + 2 * i] - glx) + smooth_l1f(ld[5 + 2 * i] - gly);
    }
    l_iou = smooth_l1f(iou[bp] - ov);
    l_posce = ce;
    cnt = 1.0f;
  }

  // wave-level reductions on the matrix unit (all lanes converged here)
  float r0 = wave_sum32(l_eiou);
  float r1 = wave_sum32(l_iou);
  float r2 = wave_sum32(l_lm);
  float r3 = wave_sum32(l_posce);
  float r4 = wave_sum32(cnt);
  if ((threadIdx.x & 31) == 0) {
    atomicAdd(&accum[0], r0);
    atomicAdd(&accum[1], r1);
    atomicAdd(&accum[2], r2);
    atomicAdd(&accum[3], r3);
    atomicAdd(&npos[b], (int)(r4 + 0.5f));
  }
}

// ---------------- hard negative mining: bitwise k-th-largest select -------
__device__ int block_count_gt(const float* __restrict__ lc, int P, float v,
                              int* scratch) {
  int local = 0;
  for (int p = threadIdx.x; p < P; p += blockDim.x) local += (lc[p] > v);
  __syncthreads();
  if (threadIdx.x == 0) *scratch = 0;
  __syncthreads();
  atomicAdd(scratch, local);
  __syncthreads();
  return *scratch;
}

__global__ void k_negmine(const float* __restrict__ loss_c,
                          const int* __restrict__ npos, float* accum, int P,
                          int T) {
  int b = blockIdx.x;
  const float* lc = loss_c + (size_t)b * P;
  long long k = 3LL * (long long)npos[b];
  if (k > (long long)(P - 1)) k = P - 1;
  if (k <= 0) return;  // uniform across block

  __shared__ int scnt;
  __shared__ float ssum;

  unsigned thr_bits;
  int c0 = block_count_gt(lc, P, 0.0f, &scnt);
  if (c0 < (int)k) {
    thr_bits = 0u;  // k-th largest is 0 (ties among zeros contribute nothing)
  } else {
    unsigned lo = 0u, hi = 0x7F800000u;  // C(lo)>=k, C(hi)=0<k
    while (hi - lo > 1u) {
      unsigned mid = lo + ((hi - lo) >> 1);
      int c = block_count_gt(lc, P, __uint_as_float(mid), &scnt);
      if (c < (int)k) hi = mid; else lo = mid;
    }
    thr_bits = hi;  // exact bits of the k-th largest element
  }
  float thr = __uint_as_float(thr_bits);

  int cg_local = 0;
  float sum_local = 0.0f;
  for (int p = threadIdx.x; p < P; p += blockDim.x) {
    float x = lc[p];
    if (x > thr) { cg_local++; sum_local += x; }
  }
  __syncthreads();
  if (threadIdx.x == 0) { scnt = 0; ssum = 0.0f; }
  __syncthreads();
  atomicAdd(&scnt, cg_local);
  atomicAdd(&ssum, sum_local);
  __syncthreads();
  if (threadIdx.x == 0) {
    // stable-argsort tie handling: (k - count_gt) elements equal to thr
    atomicAdd(&accum[4], ssum + (float)((int)k - scnt) * thr);
  }
}

// ---------------- finalize ----------------
__global__ void k_final(const float* __restrict__ accum,
                        const int* __restrict__ npos, int B,
                        float* __restrict__ out) {
  if (blockIdx.x == 0 && threadIdx.x == 0) {
    int tot = 0;
    for (int b = 0; b < B; ++b) tot += npos[b];
    float N = fmaxf((float)tot, 1.0f);
    out[0] = accum[0] / N;                  // loss_bbox_eiou / N
    out[1] = accum[1] / N;                  // loss_iou / N
    out[2] = accum[2] / (N * 5.0f);         // loss_lm / (5N)
    out[3] = (accum[3] + accum[4]) / N;     // loss_cls / N (pos ce + mined neg ce)
  }
}

static inline size_t align_up(size_t x, size_t a) { return (x + a - 1) & ~(a - 1); }

extern "C" void kernel_launch(void* const* d_in, const int* in_sizes, int n_in,
                              void* d_out, int out_size, void* d_ws, size_t ws_size,
                              hipStream_t stream) {
  const float* loc     = (const float*)d_in[0];
  const float* conf    = (const float*)d_in[1];
  const float* iou     = (const float*)d_in[2];
  const float* priors  = (const float*)d_in[3];
  const float* targets = (const float*)d_in[4];

  const int P = in_sizes[3] / 4;
  const int B = in_sizes[0] / (P * 14);
  const int T = in_sizes[4] / (B * 15);

  // workspace carve-out (~10 MB for B=32, P=25600)
  char* w = (char*)d_ws;
  float* bto = (float*)w;            w += align_up((size_t)B * P * sizeof(float), 256);
  int* bti = (int*)w;                w += align_up((size_t)B * P * sizeof(int), 256);
  float* lossc = (float*)w;          w += align_up((size_t)B * P * sizeof(float), 256);
  int* bpidx = (int*)w;              w += align_up((size_t)B * T * sizeof(int), 256);
  int* npos = (int*)w;               w += align_up((size_t)B * sizeof(int), 256);
  float* accum = (float*)w;          w += align_up(8 * sizeof(float), 256);
  (void)ws_size; (void)n_in;

  float* out = (float*)d_out;
  (void)out_size;

  k_init<<<1, 256, 0, stream>>>(accum, npos, B);

  dim3 gridBP((P + 255) / 256, B);
  k_best_truth<<<gridBP, 256, 0, stream>>>(priors, targets, bto, bti, P, T);

  dim3 gridTB(T, B);
  k_best_prior<<<gridTB, 256, 0, stream>>>(priors, targets, bpidx, P, T);

  k_scatter<<<1, B, 0, stream>>>(bpidx, bto, bti, P, T, B);

  k_loss<<<gridBP, 256, 0, stream>>>(loc, conf, iou, priors, targets, bto, bti,
                                     lossc, accum, npos, P, T);

  k_negmine<<<B, 256, 0, stream>>>(lossc, npos, accum, P, T);

  k_final<<<1, 32, 0, stream>>>(accum, npos, B, out);
}